// GHMCLoss_16329465659915
// MI455X (gfx1250) — compile-verified
//
#include <hip/hip_runtime.h>
#include <stdint.h>

#define BINS      10
#define MOMENTUM  0.9f
#define CB        19                 // channels
#define HW        (512 * 512)        // 2^18
#define NPIX      (8 * HW)           // 2,097,152 pixels
#define T1        128                // pass-1 block: 4 waves (wave32)
#define PPB       (T1 * 4)           // 512 pixels per block tile
#define RBLOCKS   2048               // reduction grid
#define RTHREADS  256

// ---------------------------------------------------------------------------
// CDNA5 async-tensor helpers. Generic pointers to __shared__ carry the LDS
// offset in their low 32 bits (aperture format) == VDST operand of
// global_load_async_to_lds_*.
// ---------------------------------------------------------------------------
__device__ __forceinline__ uint32_t lds_off_u32(const void* p) {
  return (uint32_t)(uintptr_t)p;
}

__device__ __forceinline__ void async_b128_to_lds(uint32_t lds_byte_off,
                                                  const float* gptr) {
  uint64_t ga = (uint64_t)(uintptr_t)gptr;
  asm volatile("global_load_async_to_lds_b128 %0, %1, off th:TH_LOAD_NT"
               :: "v"(lds_byte_off), "v"(ga)
               : "memory");
}

__device__ __forceinline__ void wait_async0() {
  asm volatile("s_wait_asynccnt 0" ::: "memory");
}

// searchsorted(edges, g, 'left') - 1, clipped to [0, BINS-1].
// edges = {0, .1, ..., .9, 1+1e-6}; index = (#edges strictly < g) - 1.
__device__ __forceinline__ int bucketize(float g) {
  int cnt = 0;
#pragma unroll
  for (int k = 0; k <= BINS; ++k) {
    const float e = (k == BINS) ? (1.0f + 1e-6f) : (float)k * 0.1f;
    cnt += (e < g) ? 1 : 0;
  }
  int ind = cnt - 1;
  return ind < 0 ? 0 : (ind > BINS - 1 ? BINS - 1 : ind);
}

struct Px4 {
  float ce[4];
  int   ind[4];
};

// Four consecutive pixels per thread: DMA 19 channel rows (b128 per thread)
// into LDS, read each channel ONCE as float4 into registers, do the
// max / exp-sum phases out of registers, gather p_t from LDS.
__device__ __forceinline__ void ce_ind_4px(const float* __restrict__ preds,
                                           const int*   __restrict__ target,
                                           int n0 /* 16B-aligned pixel idx */,
                                           float* tile, int tid, Px4& r) {
  const int    b    = n0 >> 18;             // HW == 2^18
  const int    rem  = n0 & (HW - 1);
  const size_t base = (size_t)b * (size_t)(CB * HW) + (size_t)rem;

  const uint32_t lb = lds_off_u32(&tile[4 * tid]);
#pragma unroll
  for (int c = 0; c < CB; ++c)
    async_b128_to_lds(lb + (uint32_t)(c * PPB * 4),
                      preds + base + (size_t)c * HW);

  const int4 t4 = *(const int4*)(target + n0);   // overlaps with async DMA
  wait_async0();

  float4 v[CB];
  float m0 = -3.402823466e+38f, m1 = m0, m2 = m0, m3 = m0;
#pragma unroll
  for (int c = 0; c < CB; ++c) {
    v[c] = *(const float4*)&tile[c * PPB + 4 * tid];   // ds_load_b128
    m0 = fmaxf(m0, v[c].x); m1 = fmaxf(m1, v[c].y);
    m2 = fmaxf(m2, v[c].z); m3 = fmaxf(m3, v[c].w);
  }
  float s0 = 0.f, s1 = 0.f, s2 = 0.f, s3 = 0.f;
#pragma unroll
  for (int c = 0; c < CB; ++c) {
    s0 += __expf(v[c].x - m0); s1 += __expf(v[c].y - m1);
    s2 += __expf(v[c].z - m2); s3 += __expf(v[c].w - m3);
  }
  const float lse0 = m0 + __logf(s0), lse1 = m1 + __logf(s1);
  const float lse2 = m2 + __logf(s2), lse3 = m3 + __logf(s3);

  const float pt0 = tile[t4.x * PPB + 4 * tid + 0];
  const float pt1 = tile[t4.y * PPB + 4 * tid + 1];
  const float pt2 = tile[t4.z * PPB + 4 * tid + 2];
  const float pt3 = tile[t4.w * PPB + 4 * tid + 3];

  r.ce[0] = lse0 - pt0; r.ind[0] = bucketize(fabsf(pt0 - 1.0f));
  r.ce[1] = lse1 - pt1; r.ind[1] = bucketize(fabsf(pt1 - 1.0f));
  r.ce[2] = lse2 - pt2; r.ind[2] = bucketize(fabsf(pt2 - 1.0f));
  r.ce[3] = lse3 - pt3; r.ind[3] = bucketize(fabsf(pt3 - 1.0f));
}

// ---------------------------------------------------------------------------
// Pass 1: single streaming read of preds (NT). LDS integer histogram
// (ds_add_u32, exact => deterministic). Stash ce (f32x4) + packed bins (u32).
// ---------------------------------------------------------------------------
template <bool STORE>
__global__ void __launch_bounds__(T1)
ghm_pass1(const float* __restrict__ preds, const int* __restrict__ target,
          unsigned int* __restrict__ g_hist,
          float* __restrict__ ce_out, uint8_t* __restrict__ ind_out) {
  __shared__ unsigned int       s_hist[BINS];
  __shared__ alignas(16) float  tile[CB * PPB];

  const int tid = threadIdx.x;
  if (tid < BINS) s_hist[tid] = 0u;
  __syncthreads();

  const int n0 = blockIdx.x * PPB + 4 * tid;
  Px4 r;
  ce_ind_4px(preds, target, n0, tile, tid, r);

#pragma unroll
  for (int j = 0; j < 4; ++j) atomicAdd(&s_hist[r.ind[j]], 1u);

  if (STORE) {
    *(float4*)(ce_out + n0) = make_float4(r.ce[0], r.ce[1], r.ce[2], r.ce[3]);
    *(uint32_t*)(ind_out + n0) =
        (uint32_t)r.ind[0] | ((uint32_t)r.ind[1] << 8) |
        ((uint32_t)r.ind[2] << 16) | ((uint32_t)r.ind[3] << 24);
  }
  __syncthreads();
  if (tid < BINS) atomicAdd(&g_hist[tid], s_hist[tid]);
}

// EMA + reciprocal weights (10 lanes of one wave).
__global__ void ghm_weights(const unsigned int* __restrict__ g_hist,
                            const float* __restrict__ acc_sum,
                            float* __restrict__ inv_acc) {
  const int i = threadIdx.x;
  if (i < BINS) {
    const float acc = MOMENTUM * acc_sum[i] +
                      (1.0f - MOMENTUM) * (float)g_hist[i];
    inv_acc[i] = 1.0f / (acc + 1e-6f);
  }
}

// Fixed-order block reduction: wave32 shfl tree, then thread 0 sums the per-wave LDS partials serially.
template <int BT>
__device__ __forceinline__ void block_reduce_store(float sum,
                                                   float* __restrict__ partials) {
  __shared__ float s_wsum[BT / 32];
#pragma unroll
  for (int off = 16; off > 0; off >>= 1) sum += __shfl_xor(sum, off, 32);
  const int lane = threadIdx.x & 31, wave = threadIdx.x >> 5;
  if (lane == 0) s_wsum[wave] = sum;
  __syncthreads();
  if (threadIdx.x == 0) {
    float t = 0.0f;
#pragma unroll
    for (int w = 0; w < BT / 32; ++w) t += s_wsum[w];
    partials[blockIdx.x] = t;
  }
}

// Pass 2a (fast path): weighted sum over stashed ce/ind (~10 MB, L2-resident).
__global__ void __launch_bounds__(RTHREADS)
ghm_reduce_stored(const float* __restrict__ ce, const uint8_t* __restrict__ ind,
                  const float* __restrict__ inv_acc, float* __restrict__ partials) {
  __shared__ float s_inv[BINS];
  if (threadIdx.x < BINS) s_inv[threadIdx.x] = inv_acc[threadIdx.x];
  __syncthreads();
  float sum = 0.0f;
  for (int n = (blockIdx.x * RTHREADS + threadIdx.x) * 4; n < NPIX;
       n += gridDim.x * RTHREADS * 4) {
    const float4   c4 = *(const float4*)(ce + n);
    const uint32_t i4 = *(const uint32_t*)(ind + n);
    sum += c4.x * s_inv[i4 & 255]         + c4.y * s_inv[(i4 >> 8) & 255] +
           c4.z * s_inv[(i4 >> 16) & 255] + c4.w * s_inv[i4 >> 24];
  }
  block_reduce_store<RTHREADS>(sum, partials);
}

// Pass 2b (fallback if workspace is tiny): recompute ce/ind (re-reads preds).
__global__ void __launch_bounds__(T1)
ghm_reduce_recompute(const float* __restrict__ preds, const int* __restrict__ target,
                     const float* __restrict__ inv_acc, float* __restrict__ partials) {
  __shared__ alignas(16) float tile[CB * PPB];
  __shared__ float             s_inv[BINS];
  if (threadIdx.x < BINS) s_inv[threadIdx.x] = inv_acc[threadIdx.x];
  __syncthreads();
  float sum = 0.0f;
  for (int base = blockIdx.x * PPB; base < NPIX; base += gridDim.x * PPB) {
    Px4 r;
    ce_ind_4px(preds, target, base + 4 * threadIdx.x, tile, threadIdx.x, r);
#pragma unroll
    for (int j = 0; j < 4; ++j) sum += r.ce[j] * s_inv[r.ind[j]];
  }
  block_reduce_store<T1>(sum, partials);
}

// Final fixed-order reduction of block partials in double, then mean.
__global__ void __launch_bounds__(256)
ghm_final(const float* __restrict__ partials, int count, float* __restrict__ out) {
  __shared__ double s[256];
  double t = 0.0;
  for (int i = threadIdx.x; i < count; i += 256) t += (double)partials[i];
  s[threadIdx.x] = t;
  __syncthreads();
  for (int off = 128; off > 0; off >>= 1) {
    if (threadIdx.x < off) s[threadIdx.x] += s[threadIdx.x + off];
    __syncthreads();
  }
  if (threadIdx.x == 0) out[0] = (float)(s[0] / (double)NPIX);
}

// ---------------------------------------------------------------------------
// Workspace layout (bytes):
//   [0,64)      : unsigned hist[10]      (memset to 0 each call)
//   [64,128)    : float inv_acc[10]
//   [128,8320)  : float partials[RBLOCKS]
//   [40960,...) : float ce[NPIX] ; uint8 ind[NPIX]   (fast path only)
// ---------------------------------------------------------------------------
extern "C" void kernel_launch(void* const* d_in, const int* in_sizes, int n_in,
                              void* d_out, int out_size, void* d_ws, size_t ws_size,
                              hipStream_t stream) {
  const float* preds   = (const float*)d_in[0];
  const int*   target  = (const int*)d_in[1];
  const float* acc_sum = (const float*)d_in[2];
  float*       out     = (float*)d_out;

  char* ws = (char*)d_ws;
  unsigned int* hist     = (unsigned int*)(ws + 0);
  float*        inv_acc  = (float*)(ws + 64);
  float*        partials = (float*)(ws + 128);
  const size_t  off_ce   = 40960;
  float*        ce_arr   = (float*)(ws + off_ce);
  uint8_t*      ind_arr  = (uint8_t*)(ws + off_ce + (size_t)NPIX * 4);
  const bool store = ws_size >= off_ce + (size_t)NPIX * 5;

  hipMemsetAsync(ws, 0, 64, stream);  // zero histogram (graph-capturable)

  if (store)
    ghm_pass1<true><<<NPIX / PPB, T1, 0, stream>>>(preds, target, hist,
                                                   ce_arr, ind_arr);
  else
    ghm_pass1<false><<<NPIX / PPB, T1, 0, stream>>>(preds, target, hist,
                                                    nullptr, nullptr);

  ghm_weights<<<1, 32, 0, stream>>>(hist, acc_sum, inv_acc);

  if (store)
    ghm_reduce_stored<<<RBLOCKS, RTHREADS, 0, stream>>>(ce_arr, ind_arr,
                                                        inv_acc, partials);
  else
    ghm_reduce_recompute<<<RBLOCKS, T1, 0, stream>>>(preds, target,
                                                     inv_acc, partials);

  ghm_final<<<1, 256, 0, stream>>>(partials, RBLOCKS, out);
}